// DepthConv_42949672961099
// MI455X (gfx1250) — compile-verified
//
#include <hip/hip_runtime.h>

// ---------------------------------------------------------------------------
// Depth-gated 3x3 conv for MI455X (gfx1250), fp32 WMMA path.
//   out[b,o,h,w] = sum_k gate[k,p] * (sum_c W[o,c,k] * x[c,p+off(k)]) + bias[o]
// Per-tap channel reduction mapped onto V_WMMA_F32_16X16X4_F32.
// ---------------------------------------------------------------------------

typedef __attribute__((ext_vector_type(2))) float v2f;
typedef __attribute__((ext_vector_type(8))) float v8f;

#define CIN   64
#define COUT  64
#define HH    128
#define WW    128
#define BATCH 8

// LDS x tile: [c=64][r=3][j=34] floats (j=0 <-> w0-1), then gate [k=9][col=32]
#define XS_C_STRIDE 102          // 3*34
#define XS_FLOATS   (64 * 102)   // 6528
#define GATE_OFF    XS_FLOATS    // 6528
#define LDS_FLOATS  (XS_FLOATS + 9 * 32)  // 6816 floats = 27264 B

// ---------------------------------------------------------------------------
// Pre-kernel: repack weight [Cout][Cin][9] into the WMMA fp32 A-operand
// layout:  pw[((mt*9 + k)*16 + cc)*32 + lane]  (float2 per lane).
//   A element (M,K): lane = M + 16*(K>=2), vgpr = K&1
//   => lane l holds (W[o, c], W[o, c+1]) with o = mt*16 + (l&15),
//      c = cc*4 + 2*(l>>4).
// Total: 4*9*16*32 float2 = 147456 bytes in d_ws.
// ---------------------------------------------------------------------------
__global__ void repack_weight_kernel(const float* __restrict__ w,
                                     float2* __restrict__ pw) {
    int id = blockIdx.x * blockDim.x + threadIdx.x;
    if (id >= 4 * 9 * 16 * 32) return;
    int lane = id & 31;
    int cc   = (id >> 5) & 15;
    int kk   = (id >> 9) % 9;
    int mt   = (id >> 9) / 9;
    int o = mt * 16 + (lane & 15);
    int c = cc * 4 + 2 * (lane >> 4);
    float2 v;
    v.x = w[(o * CIN + c) * 9 + kk];
    v.y = w[(o * CIN + c + 1) * 9 + kk];
    pw[id] = v;
}

// ---------------------------------------------------------------------------
// Main kernel: block = 256 threads = 8 wave32.
//   grid: x = W/32 (pixel groups), y = H, z = B
//   wave (mt = wave&3, tc = wave>>2) computes the 16x16 tile
//   [o = mt*16 .. +15] x [pixel col = w0 + tc*16 .. +15].
// ---------------------------------------------------------------------------
__global__ void __launch_bounds__(256)
depthconv_wmma_kernel(const float* __restrict__ x,
                      const float* __restrict__ depth,
                      const float* __restrict__ bias,
                      const float2* __restrict__ pw,
                      float* __restrict__ out) {
    __shared__ float lds[LDS_FLOATS];

    const int tid = threadIdx.x;
    const int b   = blockIdx.z;
    const int h   = blockIdx.y;
    const int w0  = blockIdx.x * 32;

    // ---- stage x halo tile (64 x 3 x 34), zero-padded at image borders ----
    const float* xb = x + (size_t)b * CIN * HH * WW;
    for (int idx = tid; idx < XS_FLOATS; idx += 256) {
        int c   = idx / XS_C_STRIDE;
        int rem = idx - c * XS_C_STRIDE;
        int r   = rem / 34;            // row h-1 .. h+1
        int j   = rem - r * 34;        // col w0-1 .. w0+32
        int gh  = h + r - 1;
        int gw  = w0 + j - 1;
        float v = 0.0f;
        if ((unsigned)gh < HH && (unsigned)gw < WW)
            v = xb[(c * HH + gh) * WW + gw];
        lds[idx] = v;
    }

    // ---- stage gate table: gate[k][col] for the 32 pixels of this block ----
    const float* db = depth + (size_t)b * HH * WW;
    for (int idx = tid; idx < 9 * 32; idx += 256) {
        int k  = idx >> 5;
        int j  = idx & 31;
        int dh = k / 3 - 1;
        int dw = k - (k / 3) * 3 - 1;
        int gh = h + dh;
        int gw = w0 + j + dw;
        float dc = db[h * WW + w0 + j];
        float dn = ((unsigned)gh < HH && (unsigned)gw < WW) ? db[gh * WW + gw]
                                                           : 0.0f;
        // OOB taps multiply zero-padded x, so gate value there is don't-care.
        lds[GATE_OFF + idx] = __expf(-fabsf(dn - dc));
    }
    __syncthreads();

    // ---- per-wave WMMA compute (uniform control flow: EXEC all ones) ----
    const int lane    = tid & 31;
    const int wave    = tid >> 5;
    const int mt      = wave & 3;       // which 16-row Cout tile
    const int tc      = wave >> 2;      // which 16-col pixel subtile
    const int halfsel = lane >> 4;      // 0: lanes 0-15, 1: lanes 16-31
    const int col     = tc * 16 + (lane & 15);   // pixel col within block

    v8f acc = {0.f, 0.f, 0.f, 0.f, 0.f, 0.f, 0.f, 0.f};

    const float2* ap0 = pw + (size_t)mt * 9 * 16 * 32 + lane;

    for (int k = 0; k < 9; ++k) {
        const int dh  = k / 3;          // 0..2 -> LDS row index r
        const int dwj = k - dh * 3;     // 0..2 ; LDS col j = col + dwj
        v8f y = {0.f, 0.f, 0.f, 0.f, 0.f, 0.f, 0.f, 0.f};

        const float2* ap = ap0 + k * 16 * 32;
        // B element (K,N): lane holds K = 2*halfsel (+1), N = lane&15.
        const float* bp = lds + (2 * halfsel) * XS_C_STRIDE + dh * 34
                              + (col + dwj);
#pragma unroll
        for (int cc = 0; cc < 16; ++cc) {
            float2 at = ap[cc * 32];          // one coalesced b64 (L2-hot)
            v2f a; a.x = at.x; a.y = at.y;
            v2f bv;                           // two ds_load_b32, imm offsets
            bv.x = bp[cc * (4 * XS_C_STRIDE)];
            bv.y = bp[cc * (4 * XS_C_STRIDE) + XS_C_STRIDE];
            y = __builtin_amdgcn_wmma_f32_16x16x4_f32(
                    false, a, false, bv, (short)0, y, false, false);
        }
        // gate is constant across the M (Cout) dimension -> scalar per lane
        float g = lds[GATE_OFF + k * 32 + col];
#pragma unroll
        for (int i = 0; i < 8; ++i) acc[i] = fmaf(g, y[i], acc[i]);
    }

    // ---- store: D element (M,N): M = r + 8*halfsel, N = lane&15 ----
    const int o0 = mt * 16 + halfsel * 8;
    float* ob = out + (size_t)b * COUT * HH * WW;
#pragma unroll
    for (int r = 0; r < 8; ++r) {
        int o = o0 + r;
        ob[((size_t)o * HH + h) * WW + w0 + col] = acc[r] + bias[o];
    }
}

// ---------------------------------------------------------------------------
extern "C" void kernel_launch(void* const* d_in, const int* in_sizes, int n_in,
                              void* d_out, int out_size, void* d_ws,
                              size_t ws_size, hipStream_t stream) {
    const float* x      = (const float*)d_in[0];
    const float* depth  = (const float*)d_in[1];
    const float* weight = (const float*)d_in[2];
    const float* bias   = (const float*)d_in[3];
    float*  out = (float*)d_out;
    float2* pw  = (float2*)d_ws;   // needs 147456 B of scratch

    // 4*9*16*32 = 18432 float2 entries -> 72 blocks of 256
    repack_weight_kernel<<<dim3(72), dim3(256), 0, stream>>>(weight, pw);

    dim3 grid(WW / 32, HH, BATCH);   // 4 x 128 x 8
    depthconv_wmma_kernel<<<grid, dim3(256), 0, stream>>>(x, depth, bias, pw,
                                                          out);
}